// GNNEncoder_26860725469580
// MI455X (gfx1250) — compile-verified
//
#include <hip/hip_runtime.h>
#include <hip/hip_bf16.h>
#include <math.h>

#define NEG_SLOPE 0.2f

typedef float v2f __attribute__((ext_vector_type(2)));
typedef float v8f __attribute__((ext_vector_type(8)));

// ---------------------------------------------------------------------------
// Dense projection: XL = X @ Wl, XR = X @ Wr using V_WMMA_F32_16X16X4_F32.
// X: [n, K] row-major f32, W: [K, 32] row-major f32, outputs [n, 32].
// Weights staged into LDS via CDNA5 async global->LDS DMA (ASYNCcnt), then
// one wave computes a 16-row tile of BOTH outputs (A regs shared by 4 accums).
// Launch with blockDim.x == 256.
// ---------------------------------------------------------------------------
template <int K>
__global__ void gat_proj_wmma(const float* __restrict__ X,
                              const float* __restrict__ Wl,
                              const float* __restrict__ Wr,
                              float* __restrict__ XL,
                              float* __restrict__ XR,
                              int n) {
  __shared__ __align__(16) float sW[2][K * 32];

  // ---- async stage Wl, Wr into LDS (16B per lane per issue) ----
  {
    const float* srcs[2] = {Wl, Wr};
#pragma unroll
    for (int m = 0; m < 2; ++m) {
      for (int byteOff = threadIdx.x * 16; byteOff < K * 128; byteOff += 256 * 16) {
        unsigned ldsAddr = (unsigned)(size_t)(&sW[m][0]) + (unsigned)byteOff;
        const void* gaddr = (const char*)srcs[m] + byteOff;
        asm volatile("global_load_async_to_lds_b128 %0, %1, off"
                     :
                     : "v"(ldsAddr), "v"(gaddr)
                     : "memory");
      }
    }
    asm volatile("s_wait_asynccnt 0" ::: "memory");
    __syncthreads();
  }

  const int lane = threadIdx.x & 31;
  const int wave = threadIdx.x >> 5;
  const int tile = blockIdx.x * (blockDim.x >> 5) + wave;
  const int row0 = tile * 16;
  if (row0 >= n) return;

  const int half  = lane >> 4;   // 0: lanes 0-15, 1: lanes 16-31
  const int l16   = lane & 15;
  const int kbase = half * 2;    // A: v0 holds K=0/2, v1 holds K=1/3 per half

  int arow = row0 + l16;
  if (arow >= n) arow = n - 1;   // clamp A loads; stores are guarded below

  v8f cl0 = {}, cl1 = {}, cr0 = {}, cr1 = {};

  for (int kk = 0; kk < K; kk += 4) {
    // A 16x4: lane = row (mod 16), half selects K pair
    v2f a;
    a.x = X[(size_t)arow * K + kk + kbase];
    a.y = X[(size_t)arow * K + kk + kbase + 1];

    // B 4x16 from LDS: lane%16 = column, half selects K pair; 2 N-tiles
    const int kr0 = (kk + kbase) * 32;
    const int kr1 = (kk + kbase + 1) * 32;
    v2f bl0, bl1, br0, br1;
    bl0.x = sW[0][kr0 + l16];      bl0.y = sW[0][kr1 + l16];
    bl1.x = sW[0][kr0 + 16 + l16]; bl1.y = sW[0][kr1 + 16 + l16];
    br0.x = sW[1][kr0 + l16];      br0.y = sW[1][kr1 + l16];
    br1.x = sW[1][kr0 + 16 + l16]; br1.y = sW[1][kr1 + 16 + l16];

    cl0 = __builtin_amdgcn_wmma_f32_16x16x4_f32(false, a, false, bl0, (short)0, cl0, false, false);
    cl1 = __builtin_amdgcn_wmma_f32_16x16x4_f32(false, a, false, bl1, (short)0, cl1, false, false);
    cr0 = __builtin_amdgcn_wmma_f32_16x16x4_f32(false, a, false, br0, (short)0, cr0, false, false);
    cr1 = __builtin_amdgcn_wmma_f32_16x16x4_f32(false, a, false, br1, (short)0, cr1, false, false);
  }

  // C/D layout: VGPR r -> M = r + half*8, N = lane%16 (+ ntile*16)
#pragma unroll
  for (int r = 0; r < 8; ++r) {
    const int m = row0 + r + half * 8;
    if (m < n) {
      XL[(size_t)m * 32 + l16]      = cl0[r];
      XL[(size_t)m * 32 + 16 + l16] = cl1[r];
      XR[(size_t)m * 32 + l16]      = cr0[r];
      XR[(size_t)m * 32 + 16 + l16] = cr1[r];
    }
  }
}

// ---------------------------------------------------------------------------
// Zero/neg-inf init of per-layer accumulators.
// ---------------------------------------------------------------------------
__global__ void gat_init(float* __restrict__ acc, float* __restrict__ emax,
                         float* __restrict__ denom, int n) {
  const int idx = blockIdx.x * blockDim.x + threadIdx.x;
  if (idx < n * 32) acc[idx] = 0.0f;
  if (idx < n) {
    emax[idx]  = -INFINITY;
    denom[idx] = 0.0f;
  }
}

// ---------------------------------------------------------------------------
// Edge pass 1: e = att . leaky_relu(xl[src] + xr[dst]); segment max via
// native CDNA5 float atomic max (global_atomic_max_num_f32).
// ---------------------------------------------------------------------------
__global__ void gat_edge_logits(const int* __restrict__ src,
                                const int* __restrict__ dst,
                                const float* __restrict__ XL,
                                const float* __restrict__ XR,
                                const float* __restrict__ att,
                                float* __restrict__ elog,
                                float* __restrict__ emax, int ne) {
  const int e = blockIdx.x * blockDim.x + threadIdx.x;
  if (e >= ne) return;
  const int s = src[e];
  const int d = dst[e];
  const float4* pl = (const float4*)(XL + (size_t)s * 32);
  const float4* pr = (const float4*)(XR + (size_t)d * 32);
  const float4* pa = (const float4*)att;
  float logit = 0.0f;
#pragma unroll
  for (int q = 0; q < 8; ++q) {
    const float4 l = pl[q], r = pr[q], a = pa[q];
    float m;
    m = l.x + r.x; m = m > 0.0f ? m : m * NEG_SLOPE; logit += m * a.x;
    m = l.y + r.y; m = m > 0.0f ? m : m * NEG_SLOPE; logit += m * a.y;
    m = l.z + r.z; m = m > 0.0f ? m : m * NEG_SLOPE; logit += m * a.z;
    m = l.w + r.w; m = m > 0.0f ? m : m * NEG_SLOPE; logit += m * a.w;
  }
  elog[e] = logit;
  float* addr = emax + d;
  asm volatile("global_atomic_max_num_f32 %0, %1, off scope:SCOPE_DEV"
               :
               : "v"(addr), "v"(logit)
               : "memory");
}

// ---------------------------------------------------------------------------
// Edge pass 2 (fused): p = exp(e - emax[dst]); denom[dst] += p;
// acc[dst,:] += p * xl[src,:]. (alpha division deferred to finalize.)
// ---------------------------------------------------------------------------
__global__ void gat_edge_accum(const int* __restrict__ src,
                               const int* __restrict__ dst,
                               const float* __restrict__ XL,
                               const float* __restrict__ elog,
                               const float* __restrict__ emax,
                               float* __restrict__ denom,
                               float* __restrict__ acc, int ne) {
  const int e = blockIdx.x * blockDim.x + threadIdx.x;
  if (e >= ne) return;
  const int s = src[e];
  const int d = dst[e];
  const float p = __expf(elog[e] - emax[d]);
  atomicAdd(denom + d, p);
  const float* pl = XL + (size_t)s * 32;
  float* pa = acc + (size_t)d * 32;
#pragma unroll
  for (int h = 0; h < 32; ++h) {
    atomicAdd(pa + h, p * pl[h]);
  }
}

// ---------------------------------------------------------------------------
// Finalize: out = acc/denom + b (empty segments -> b), optional ReLU.
// ---------------------------------------------------------------------------
__global__ void gat_finalize(const float* __restrict__ acc,
                             const float* __restrict__ denom,
                             const float* __restrict__ bias,
                             float* __restrict__ out, int n, int do_relu) {
  const int idx = blockIdx.x * blockDim.x + threadIdx.x;
  if (idx >= n * 32) return;
  const int i = idx >> 5;
  const int h = idx & 31;
  const float den = denom[i];
  float v = den > 0.0f ? acc[idx] / den : 0.0f;
  v += bias[h];
  if (do_relu) v = v > 0.0f ? v : 0.0f;
  out[idx] = v;
}

// ---------------------------------------------------------------------------
extern "C" void kernel_launch(void* const* d_in, const int* in_sizes, int n_in,
                              void* d_out, int out_size, void* d_ws, size_t ws_size,
                              hipStream_t stream) {
  const float* x    = (const float*)d_in[0];
  const int*   ei   = (const int*)d_in[1];
  const float* Wl1  = (const float*)d_in[2];
  const float* Wr1  = (const float*)d_in[3];
  const float* att1 = (const float*)d_in[4];
  const float* b1   = (const float*)d_in[5];
  const float* Wl2  = (const float*)d_in[6];
  const float* Wr2  = (const float*)d_in[7];
  const float* att2 = (const float*)d_in[8];
  const float* b2   = (const float*)d_in[9];

  const int D  = 64;
  const int ne = in_sizes[1] / 2;        // E
  const int n  = in_sizes[0] / D;        // N
  const int*   src = ei;
  const int*   dst = ei + ne;

  // workspace layout (floats)
  const size_t N32 = (size_t)n * 32;
  float* xl    = (float*)d_ws;
  float* xr    = xl + N32;
  float* hbuf  = xr + N32;
  float* accb  = hbuf + N32;
  float* elog  = accb + N32;
  float* emax  = elog + (size_t)ne;
  float* denom = emax + (size_t)n;

  const int BLK = 256;
  const int wavesPerBlk = BLK / 32;
  const int tiles       = (n + 15) / 16;
  const int projBlocks  = (tiles + wavesPerBlk - 1) / wavesPerBlk;
  const int edgeBlocks  = (ne + BLK - 1) / BLK;
  const int nodeBlocks  = (n * 32 + BLK - 1) / BLK;
  float* out = (float*)d_out;

  // ---- layer 1: D=64 -> H=32
  gat_proj_wmma<64><<<projBlocks, BLK, 0, stream>>>(x, Wl1, Wr1, xl, xr, n);
  gat_init<<<nodeBlocks, BLK, 0, stream>>>(accb, emax, denom, n);
  gat_edge_logits<<<edgeBlocks, BLK, 0, stream>>>(src, dst, xl, xr, att1, elog, emax, ne);
  gat_edge_accum<<<edgeBlocks, BLK, 0, stream>>>(src, dst, xl, elog, emax, denom, accb, ne);
  gat_finalize<<<nodeBlocks, BLK, 0, stream>>>(accb, denom, b1, hbuf, n, /*relu=*/1);

  // ---- layer 2: H=32 -> H=32
  gat_proj_wmma<32><<<projBlocks, BLK, 0, stream>>>(hbuf, Wl2, Wr2, xl, xr, n);
  gat_init<<<nodeBlocks, BLK, 0, stream>>>(accb, emax, denom, n);
  gat_edge_logits<<<edgeBlocks, BLK, 0, stream>>>(src, dst, xl, xr, att2, elog, emax, ne);
  gat_edge_accum<<<edgeBlocks, BLK, 0, stream>>>(src, dst, xl, elog, emax, denom, accb, ne);
  gat_finalize<<<nodeBlocks, BLK, 0, stream>>>(accb, denom, b2, out, n, /*relu=*/0);
}